// QNet_58875411694186
// MI455X (gfx1250) — compile-verified
//
#include <hip/hip_runtime.h>

typedef __attribute__((ext_vector_type(2))) float v2f;
typedef __attribute__((ext_vector_type(8))) float v8f;

constexpr int NQ    = 12;
constexpr int DIM   = 4096;
constexpr int DEPTH = 6;
constexpr float PI_F = 3.14159265358979323846f;

__device__ __forceinline__ v8f wmma16x16x4(v2f a, v2f b, v8f c) {
    // D = A(16x4) * B(4x16) + C, full f32. 8-arg form:
    // (neg_a, A, neg_b, B, c_mod, C, reuse_a, reuse_b)
    return __builtin_amdgcn_wmma_f32_16x16x4_f32(false, a, false, b, (short)0, c,
                                                 false, false);
}

__global__ __launch_bounds__(256)
void qnet_wmma_kernel(const float* __restrict__ x,   // [256,12]
                      const float* __restrict__ th,  // [6,12]
                      float* __restrict__ out)       // [256,3]
{
    __shared__ float st[DIM];        // 16 KB real state (imag is identically 0)
    __shared__ float Umat[3][256];   // Ua/Ub/Uc 16x16; reused for final reduction
    __shared__ float lc[NQ], ls[NQ];

    const int tid  = threadIdx.x;      // 0..255
    const int b    = blockIdx.x;       // batch element
    const int lane = tid & 31;
    const int wave = tid >> 5;         // 0..7
    const int hi   = lane >> 4;        // 0: lanes 0-15, 1: lanes 16-31
    const int l    = lane & 15;

    // ---------------- encoder: product state from RY(pi*x) ----------------
    if (tid < NQ) {
        float h = x[b * NQ + tid] * (PI_F * 0.5f);
        lc[tid] = cosf(h);
        ls[tid] = sinf(h);
    }
    __syncthreads();
    {
        const int i0 = tid * 16;
        #pragma unroll
        for (int j = 0; j < 16; ++j) {
            int i = i0 + j;
            float p = 1.0f;
            #pragma unroll
            for (int q = 0; q < NQ; ++q)
                p *= ((i >> (11 - q)) & 1) ? ls[q] : lc[q];
            st[i] = p;
        }
    }
    __syncthreads();

    // ---------------- ansatz layers ----------------
    for (int d = 0; d < DEPTH; ++d) {
        if (tid < NQ) {
            float h = th[d * NQ + tid] * 0.5f;
            lc[tid] = cosf(h);
            ls[tid] = sinf(h);
        }
        __syncthreads();

        // Build Ua/Ub/Uc = kron of four RY 2x2 gates (qubit 4g+j <-> bit 3-j)
        {
            const int r = tid >> 4, k = tid & 15;
            #pragma unroll
            for (int g = 0; g < 3; ++g) {
                float p = 1.0f;
                #pragma unroll
                for (int j = 0; j < 4; ++j) {
                    int q  = 4 * g + j;
                    int rb = (r >> (3 - j)) & 1;
                    int kb = (k >> (3 - j)) & 1;
                    float cc = lc[q], ss = ls[q];
                    // RY = [[c,-s],[s,c]]
                    p *= rb ? (kb ? cc : ss) : (kb ? -ss : cc);
                }
                Umat[g][r * 16 + k] = p;
            }
        }
        __syncthreads();

        // ---- stage A: contract top nibble: S'(16x256) = Ua @ S ----
        #pragma unroll
        for (int t = 0; t < 2; ++t) {
            const int n0 = (2 * wave + t) * 16;   // column tile
            v8f acc{};
            #pragma unroll
            for (int kk = 0; kk < 4; ++kk) {
                const int k0 = 4 * kk + 2 * hi;
                v2f a, bb;
                a.x  = Umat[0][l * 16 + k0];
                a.y  = Umat[0][l * 16 + k0 + 1];
                bb.x = st[(k0    ) * 256 + n0 + l];
                bb.y = st[(k0 + 1) * 256 + n0 + l];
                acc = wmma16x16x4(a, bb, acc);
            }
            #pragma unroll
            for (int v = 0; v < 8; ++v)
                st[(v + 8 * hi) * 256 + n0 + l] = acc[v];
        }
        __syncthreads();

        // ---- stage B: contract middle nibble within each top-nibble block ----
        #pragma unroll
        for (int t = 0; t < 2; ++t) {
            const int base = (2 * wave + t) * 256;  // one a-block = 16x16
            v8f acc{};
            #pragma unroll
            for (int kk = 0; kk < 4; ++kk) {
                const int k0 = 4 * kk + 2 * hi;
                v2f a, bb;
                a.x  = Umat[1][l * 16 + k0];
                a.y  = Umat[1][l * 16 + k0 + 1];
                bb.x = st[base + (k0    ) * 16 + l];
                bb.y = st[base + (k0 + 1) * 16 + l];
                acc = wmma16x16x4(a, bb, acc);
            }
            #pragma unroll
            for (int v = 0; v < 8; ++v)
                st[base + (v + 8 * hi) * 16 + l] = acc[v];
        }
        __syncthreads();

        // ---- stage C: contract low nibble: D'(256x16) = D @ Uc^T ----
        // (A-layout load of Uc == B-operand of Uc^T under the symmetric layout)
        #pragma unroll
        for (int t = 0; t < 2; ++t) {
            const int r0 = (2 * wave + t) * 16;   // row tile
            v8f acc{};
            #pragma unroll
            for (int kk = 0; kk < 4; ++kk) {
                const int k0 = 4 * kk + 2 * hi;
                v2f a, bb;
                a.x  = st[(r0 + l) * 16 + k0];
                a.y  = st[(r0 + l) * 16 + k0 + 1];
                bb.x = Umat[2][l * 16 + k0];
                bb.y = Umat[2][l * 16 + k0 + 1];
                acc = wmma16x16x4(a, bb, acc);
            }
            #pragma unroll
            for (int v = 0; v < 8; ++v)
                st[(r0 + v + 8 * hi) * 16 + l] = acc[v];
        }
        __syncthreads();

        // ---- CNOT ladder == binary -> Gray-code gather: s[i] = s[i ^ (i>>1)] ----
        float tmp[16];
        {
            const int i0 = tid * 16;
            #pragma unroll
            for (int j = 0; j < 16; ++j) {
                int i = i0 + j;
                tmp[j] = st[i ^ (i >> 1)];
            }
        }
        __syncthreads();
        {
            const int i0 = tid * 16;
            #pragma unroll
            for (int j = 0; j < 16; ++j)
                st[i0 + j] = tmp[j];
        }
        __syncthreads();
    }

    // ---------------- observer: <Z> on qubits 0,1,2 (bits 11,10,9) ----------------
    {
        float p0 = 0.f, p1 = 0.f, p2 = 0.f;
        const int i0 = tid * 16;
        #pragma unroll
        for (int j = 0; j < 16; ++j) {
            int i = i0 + j;
            float v  = st[i];
            float pv = v * v;
            p0 += ((i >> 11) & 1) ? -pv : pv;
            p1 += ((i >> 10) & 1) ? -pv : pv;
            p2 += ((i >>  9) & 1) ? -pv : pv;
        }
        Umat[0][tid] = p0;
        Umat[1][tid] = p1;
        Umat[2][tid] = p2;
    }
    __syncthreads();
    for (int s = 128; s > 0; s >>= 1) {
        if (tid < s) {
            Umat[0][tid] += Umat[0][tid + s];
            Umat[1][tid] += Umat[1][tid + s];
            Umat[2][tid] += Umat[2][tid + s];
        }
        __syncthreads();
    }
    if (tid < 3)
        out[b * 3 + tid] = Umat[tid][0];
}

extern "C" void kernel_launch(void* const* d_in, const int* in_sizes, int n_in,
                              void* d_out, int out_size, void* d_ws, size_t ws_size,
                              hipStream_t stream) {
    const float* x  = (const float*)d_in[0];   // input_tensor [256,12] f32
    const float* th = (const float*)d_in[1];   // thetas       [6,12]  f32
    float* out = (float*)d_out;                // [256,3] f32
    const int batch = in_sizes[0] / NQ;        // 256
    qnet_wmma_kernel<<<dim3(batch), dim3(256), 0, stream>>>(x, th, out);
}